// LightGCN_63376537420313
// MI455X (gfx1250) — compile-verified
//
#include <hip/hip_runtime.h>

#define NUM_USERS 100000
#define NUM_ITEMS 50000
#define N_NODES   150000
#define EMB_DIM   64
#define N_EDGES   4000000
#define BATCH     4096
#define N_LAYERS  3

typedef __attribute__((ext_vector_type(2))) float v2f;
typedef __attribute__((ext_vector_type(8))) float v8f;

// ---------------------------------------------------------------------------
// init: cur = acc = concat(user_emb, item_emb); one float4 per thread
// ---------------------------------------------------------------------------
__global__ void lgcn_init(const float* __restrict__ user_emb,
                          const float* __restrict__ item_emb,
                          float* __restrict__ cur,
                          float* __restrict__ acc) {
    size_t i = (size_t)blockIdx.x * blockDim.x + threadIdx.x;   // float4 index
    const size_t total_f4 = (size_t)N_NODES * EMB_DIM / 4;
    if (i >= total_f4) return;
    const size_t user_f4 = (size_t)NUM_USERS * EMB_DIM / 4;
    float4 v = (i < user_f4) ? ((const float4*)user_emb)[i]
                             : ((const float4*)item_emb)[i - user_f4];
    ((float4*)cur)[i] = v;
    ((float4*)acc)[i] = v;
}

// ---------------------------------------------------------------------------
// zero the scatter target
// ---------------------------------------------------------------------------
__global__ void lgcn_zero(float* __restrict__ buf) {
    size_t i = (size_t)blockIdx.x * blockDim.x + threadIdx.x;
    const size_t total_f4 = (size_t)N_NODES * EMB_DIM / 4;
    if (i >= total_f4) return;
    ((float4*)buf)[i] = make_float4(0.f, 0.f, 0.f, 0.f);
}

// ---------------------------------------------------------------------------
// SpMM scatter: y[row[e]] += vals[e] * x[col[e]]
// 16 lanes per edge, float4 per lane -> contiguous 256B gather per edge,
// hardware f32 atomics for the scatter. x/y live in L2 (38.4 MB each).
// ---------------------------------------------------------------------------
__global__ void lgcn_spmm(const float* __restrict__ vals,
                          const int*   __restrict__ row,
                          const int*   __restrict__ col,
                          const float* __restrict__ x,
                          float*       __restrict__ y) {
    size_t tid = (size_t)blockIdx.x * blockDim.x + threadIdx.x;
    size_t e   = tid >> 4;                 // edge index (16 lanes/edge)
    if (e >= (size_t)N_EDGES) return;
    int   sub = (int)(tid & 15);           // which float4 of the 64-dim row
    int   c   = col[e];
    int   r   = row[e];
    float v   = vals[e];

    const float4 xv = *(const float4*)(x + (size_t)c * EMB_DIM + sub * 4);
    float* dst = y + (size_t)r * EMB_DIM + sub * 4;
    unsafeAtomicAdd(dst + 0, v * xv.x);    // global_atomic_add_f32 (no return)
    unsafeAtomicAdd(dst + 1, v * xv.y);
    unsafeAtomicAdd(dst + 2, v * xv.z);
    unsafeAtomicAdd(dst + 3, v * xv.w);
}

// ---------------------------------------------------------------------------
// acc += layer output
// ---------------------------------------------------------------------------
__global__ void lgcn_accum(const float* __restrict__ src,
                           float* __restrict__ acc) {
    size_t i = (size_t)blockIdx.x * blockDim.x + threadIdx.x;
    const size_t total_f4 = (size_t)N_NODES * EMB_DIM / 4;
    if (i >= total_f4) return;
    float4 a = ((const float4*)acc)[i];
    float4 s = ((const float4*)src)[i];
    a.x += s.x; a.y += s.y; a.z += s.z; a.w += s.w;
    ((float4*)acc)[i] = a;
}

// ---------------------------------------------------------------------------
// Scoring via V_WMMA_F32_16X16X4_F32.
// One wave32 per 16-row tile of the batch. For the tile:
//   D = Ue(16xK) x Pe^T(Kx16)  -> scores are diag(D).
// A-frag layout (16x4 f32): lanes 0-15 carry K=0,1; lanes 16-31 carry K=2,3
// for row M = lane&15. B-frag (4x16): identical pattern with N = lane&15.
// C/D layout: VGPR r holds M=r (lanes 0-15, N=lane) and M=r+8 (lanes 16-31,
// N=lane-16) -> diag element m sits at lane m (m<8, comp m) or lane 16+m
// (m>=8, comp m-8).
// ---------------------------------------------------------------------------
__global__ void lgcn_score(const float* __restrict__ acc,
                           const int*   __restrict__ users,
                           const int*   __restrict__ pos_items,
                           const int*   __restrict__ neg_items,
                           float*       __restrict__ out) {
    int tile = blockIdx.x;            // BATCH/16 tiles
    int lane = threadIdx.x;           // 0..31 (one wave)
    int m    = lane & 15;
    int koff = (lane < 16) ? 0 : 2;

    int b = tile * 16 + m;
    const float* urow = acc + (size_t)users[b] * EMB_DIM;
    const float* prow = acc + ((size_t)pos_items[b] + NUM_USERS) * EMB_DIM;
    const float* nrow = acc + ((size_t)neg_items[b] + NUM_USERS) * EMB_DIM;

    v8f cpos = {};
    v8f cneg = {};
#pragma unroll
    for (int k = 0; k < EMB_DIM; k += 4) {
        v2f a  = { urow[k + koff], urow[k + koff + 1] };
        v2f bp = { prow[k + koff], prow[k + koff + 1] };
        v2f bn = { nrow[k + koff], nrow[k + koff + 1] };
        cpos = __builtin_amdgcn_wmma_f32_16x16x4_f32(
                   false, a, false, bp, (short)0, cpos, false, false);
        cneg = __builtin_amdgcn_wmma_f32_16x16x4_f32(
                   false, a, false, bn, (short)0, cneg, false, false);
    }

    // diagonal extraction (EXEC divergence only after all WMMAs)
    bool lo = (lane < 8);
    bool hi = (lane >= 24);
    if (lo || hi) {
        int comp = lo ? lane : (lane - 24);   // which C/D component
        int rowm = lo ? lane : (lane - 16);   // which batch row in tile
        float p = 0.f, n = 0.f;
#pragma unroll
        for (int i = 0; i < 8; ++i) {
            if (comp == i) { p = cpos[i]; n = cneg[i]; }
        }
        const float s = 1.0f / ((N_LAYERS + 1) * (N_LAYERS + 1));  // (acc/4)·(acc/4)
        out[tile * 16 + rowm]         = p * s;
        out[BATCH + tile * 16 + rowm] = n * s;
    }
}

// ---------------------------------------------------------------------------
// launcher
// ---------------------------------------------------------------------------
extern "C" void kernel_launch(void* const* d_in, const int* in_sizes, int n_in,
                              void* d_out, int out_size, void* d_ws, size_t ws_size,
                              hipStream_t stream) {
    const float* user_emb  = (const float*)d_in[0];
    const float* item_emb  = (const float*)d_in[1];
    const float* vals      = (const float*)d_in[2];
    const int*   row       = (const int*)d_in[3];
    const int*   col       = (const int*)d_in[4];
    const int*   users     = (const int*)d_in[5];
    const int*   pos_items = (const int*)d_in[6];
    const int*   neg_items = (const int*)d_in[7];
    float*       out       = (float*)d_out;

    const size_t nfloats = (size_t)N_NODES * EMB_DIM;
    float* bufA = (float*)d_ws;
    float* bufB = bufA + nfloats;
    float* acc  = bufB + nfloats;

    const size_t f4      = nfloats / 4;
    const int    blk     = 256;
    const int    gridEl  = (int)((f4 + blk - 1) / blk);

    lgcn_init<<<gridEl, blk, 0, stream>>>(user_emb, item_emb, bufA, acc);

    float* cur = bufA;
    float* nxt = bufB;
    const size_t spmm_threads = (size_t)N_EDGES * 16;
    const int    gridSp = (int)((spmm_threads + blk - 1) / blk);

    for (int l = 0; l < N_LAYERS; ++l) {
        lgcn_zero <<<gridEl, blk, 0, stream>>>(nxt);
        lgcn_spmm <<<gridSp, blk, 0, stream>>>(vals, row, col, cur, nxt);
        lgcn_accum<<<gridEl, blk, 0, stream>>>(nxt, acc);
        float* t = cur; cur = nxt; nxt = t;
    }

    lgcn_score<<<BATCH / 16, 32, 0, stream>>>(acc, users, pos_items, neg_items, out);
}